// EMAVectorQuantizer_26439818674878
// MI455X (gfx1250) — compile-verified
//
#include <hip/hip_runtime.h>
#include <hip/hip_bf16.h>

// Problem constants
#define D_DIM   256
#define K_TOK   8192
#define N_TOK   16384      // 16*32*32
#define OUT_MAIN_SZ 4194304
#define DECAY   0.99f
#define ONEMD   0.01f
#define EPSV    1e-5f
#define KCHUNKS 4          // K-split for occupancy: 512 blocks, 4096 waves
#define TILES_PER_CHUNK (K_TOK / 16 / KCHUNKS)   // 128

typedef __attribute__((ext_vector_type(2))) float v2f;
typedef __attribute__((ext_vector_type(8))) float v8f;

// ---- optional gfx1250 async global->LDS path (ASYNCcnt), with sync fallback ----
#if defined(__has_builtin)
#  if __has_builtin(__builtin_amdgcn_global_load_async_to_lds_b32)
#    define HAVE_ASYNC_LDS 1
#  endif
#endif
#ifndef HAVE_ASYNC_LDS
#  define HAVE_ASYNC_LDS 0
#endif

#if HAVE_ASYNC_LDS
typedef __attribute__((address_space(1))) int gint_t;   // prints as "__device__ int" (probe-learned)
typedef __attribute__((address_space(3))) int lint_t;   // LDS
static __device__ __forceinline__ void async_copy_b32(float* ldst, const float* gsrc) {
  // flat LDS pointers carry the LDS byte offset in the low 32 bits (aperture mapping),
  // AS1 has the same representation as flat for global addresses.
  __builtin_amdgcn_global_load_async_to_lds_b32(
      (gint_t*)(uintptr_t)gsrc, (lint_t*)(uintptr_t)ldst, /*offset=*/0, /*cpol=*/0);
}
static __device__ __forceinline__ void wait_async_lds() {
#if __has_builtin(__builtin_amdgcn_s_wait_asynccnt)
  __builtin_amdgcn_s_wait_asynccnt(0);
#else
  asm volatile("s_wait_asynccnt 0" ::: "memory");
#endif
}
#endif

// D = A(16x4 f32) * B(4x16 f32) + C(16x16 f32)
static __device__ __forceinline__ v8f wmma_f32(v2f a, v2f b, v8f c) {
  return __builtin_amdgcn_wmma_f32_16x16x4_f32(
      /*neg_a=*/false, a, /*neg_b=*/false, b,
      /*c_mod=*/(short)0, c, /*reuse_a=*/false, /*reuse_b=*/false);
}

// -------- ||e_k||^2 --------
__global__ void e2_kernel(const float* __restrict__ embed, float* __restrict__ e2) {
  int k = blockIdx.x * 256 + threadIdx.x;
  float s = 0.f;
  for (int d = 0; d < D_DIM; ++d) {
    float v = embed[(size_t)d * K_TOK + k];
    s += v * v;
  }
  e2[k] = s;
}

// -------- fused GEMM (WMMA f32 16x16x4) + row argmin over one K chunk --------
// grid: (128, KCHUNKS) blocks x 256 threads. Block covers 128 tokens (8 waves x 16)
// and 2048 codebook columns; partial (val,idx) per token go to workspace.
#define LDSB_PITCH 260   // 256 + 4 pad -> ds_load_b64 banks 4c+{0..3}, conflict-free
__global__ __launch_bounds__(256, 1)
void vq_argmin_kernel(const float* __restrict__ x,
                      const float* __restrict__ embed,
                      const float* __restrict__ e2,
                      float* __restrict__ pv_out,   // [KCHUNKS][N_TOK]
                      int* __restrict__ pi_out) {   // [KCHUNKS][N_TOK]
  __shared__ __align__(16) float ldsB[2][16 * LDSB_PITCH]; // 2 x 16.6KB, transposed [col][d]

  const int tid   = threadIdx.x;
  const int wave  = tid >> 5;
  const int lane  = tid & 31;
  const int col   = lane & 15;   // N index within 16x16 tile
  const int hi    = lane >> 4;   // half-wave select
  const int chunk = blockIdx.y;
  const int tile0 = chunk * TILES_PER_CHUNK;
  const int m0    = blockIdx.x * 128 + wave * 16;
  const int row   = m0 + col;    // A: M = lane%16

  // Preload A panel (16 tokens x 256 d) into VGPRs: 64 K-chunks of 4.
  // ISA 16x4 f32 A layout: VGPR0 holds K=2*hi, VGPR1 holds K=2*hi+1.
  v2f a[64];
  {
    const float* xrow = x + (size_t)row * D_DIM + 2 * hi;
#pragma unroll
    for (int i = 0; i < 64; ++i) a[i] = *(const v2f*)(xrow + 4 * i);
  }

  float bestVal[8];
  int   bestIdx[8];
#pragma unroll
  for (int v = 0; v < 8; ++v) { bestVal[v] = 3.4e38f; bestIdx[v] = 0; }

  const int fc = tid & 15;    // fill column
  const int fd = tid >> 4;    // fill d-row base

  // Fill B tile 0 of this chunk: embed[d][n0+c] -> ldsB[0][c*PITCH + d]
  {
    const int n0 = tile0 * 16;
#pragma unroll
    for (int it = 0; it < 16; ++it) {
      int d = it * 16 + fd;
#if HAVE_ASYNC_LDS
      async_copy_b32(&ldsB[0][fc * LDSB_PITCH + d], &embed[(size_t)d * K_TOK + n0 + fc]);
#else
      ldsB[0][fc * LDSB_PITCH + d] = embed[(size_t)d * K_TOK + n0 + fc];
#endif
    }
  }
#if HAVE_ASYNC_LDS
  wait_async_lds();
#endif
  __syncthreads();

  for (int kt = 0; kt < TILES_PER_CHUNK; ++kt) {
    const int n0  = (tile0 + kt) * 16;
    const int cur = kt & 1;

    // prefetch next B tile into the other buffer (in flight during the WMMAs)
    if (kt + 1 < TILES_PER_CHUNK) {
      const int nn0 = n0 + 16;
      const int nxt = cur ^ 1;
#pragma unroll
      for (int it = 0; it < 16; ++it) {
        int d = it * 16 + fd;
#if HAVE_ASYNC_LDS
        async_copy_b32(&ldsB[nxt][fc * LDSB_PITCH + d], &embed[(size_t)d * K_TOK + nn0 + fc]);
#else
        ldsB[nxt][fc * LDSB_PITCH + d] = embed[(size_t)d * K_TOK + nn0 + fc];
#endif
      }
    }

    // 64 x (ds_load_b64 + v_wmma_f32_16x16x4_f32) over D=256
    v8f acc = {};
    const float* bp = &ldsB[cur][col * LDSB_PITCH + 2 * hi];
#pragma unroll
    for (int i = 0; i < 64; ++i) {
      v2f b = *(const v2f*)(bp + 4 * i);   // B VGPR0=K 2*hi, VGPR1=K 2*hi+1
      acc = wmma_f32(a[i], b, acc);
    }

    // score = ||e||^2 - 2 x.e  (||x||^2 constant per row, drop for argmin)
    const float e2v = e2[n0 + col];
    const int   kcol = n0 + col;
#pragma unroll
    for (int v = 0; v < 8; ++v) {
      float s = e2v - 2.0f * acc[v];
      if (s < bestVal[v]) { bestVal[v] = s; bestIdx[v] = kcol; }
    }
#if HAVE_ASYNC_LDS
    wait_async_lds();
#endif
    __syncthreads();
  }

  // Reduce argmin across the 16 lanes holding each row (C: M = v + 8*hi, N = lane%16)
#pragma unroll
  for (int v = 0; v < 8; ++v) {
    float bv = bestVal[v];
    int   bi = bestIdx[v];
#pragma unroll
    for (int m = 1; m < 16; m <<= 1) {
      float ov = __shfl_xor(bv, m, 32);
      int   oi = __shfl_xor(bi, m, 32);
      if (ov < bv || (ov == bv && oi < bi)) { bv = ov; bi = oi; }
    }
    if (col == 0) {
      const int mrow = m0 + v + 8 * hi;
      pv_out[chunk * N_TOK + mrow] = bv;
      pi_out[chunk * N_TOK + mrow] = bi;
    }
  }
}

// -------- combine per-chunk argmin candidates (ascending chunk = ascending k) --------
__global__ void combine_kernel(const float* __restrict__ pv, const int* __restrict__ pi,
                               int* __restrict__ idx_out) {
  const int n = blockIdx.x * 256 + threadIdx.x;
  float bv = pv[n];
  int   bi = pi[n];
  for (int c = 1; c < KCHUNKS; ++c) {
    float ov = pv[c * N_TOK + n];
    int   oi = pi[c * N_TOK + n];
    if (ov < bv || (ov == bv && oi < bi)) { bv = ov; bi = oi; }
  }
  idx_out[n] = bi;
}

// -------- embed_ind as float --------
__global__ void ind_kernel(const int* __restrict__ idx, float* __restrict__ out_ind) {
  int n = blockIdx.x * 256 + threadIdx.x;
  out_ind[n] = (float)idx[n];
}

// -------- quantize gather + [B,D,H,W] transpose + per-block MSE partials --------
__global__ __launch_bounds__(256)
void quant_out_kernel(const float* __restrict__ x, const float* __restrict__ embed,
                      const int* __restrict__ idx, float* __restrict__ out,
                      float* __restrict__ dpart) {
  __shared__ float sm[256];
  const int t = threadIdx.x;
  const size_t o = (size_t)blockIdx.x * 256 + t;  // out index [b][d][h][w]
  const int w = (int)(o & 31);
  const int h = (int)((o >> 5) & 31);
  const int d = (int)((o >> 10) & 255);
  const int b = (int)(o >> 18);
  const int n = b * 1024 + h * 32 + w;
  const int k = idx[n];
  const float q  = embed[(size_t)d * K_TOK + k];
  const float xv = x[(size_t)n * D_DIM + d];
  const float dv = q - xv;          // stop_gradient(quantize - input)
  out[o] = xv + dv;                 // straight-through value, fp-exact as reference
  sm[t] = dv * dv;
  __syncthreads();
  for (int s = 128; s > 0; s >>= 1) {
    if (t < s) sm[t] += sm[t + s];
    __syncthreads();
  }
  if (t == 0) dpart[blockIdx.x] = sm[0];
}

__global__ void diff_reduce_kernel(const float* __restrict__ dpart, float* __restrict__ out_diff) {
  __shared__ float sm[256];
  const int t = threadIdx.x;
  float s = 0.f;
  for (int i = t; i < 16384; i += 256) s += dpart[i];   // fixed order -> deterministic
  sm[t] = s;
  __syncthreads();
  for (int k = 128; k > 0; k >>= 1) {
    if (t < k) sm[t] += sm[t + k];
    __syncthreads();
  }
  if (t == 0) out_diff[0] = sm[0] / (float)OUT_MAIN_SZ;
}

// -------- deterministic segment-sum + histogram --------
// thread = (k, c) ; k = blockIdx*64 + t%64, c = t/64 owns d in [c*64, c*64+64)
// Writes raw embed_sum[d][k] into the new_embed_avg output slot (finalized later in place).
__global__ __launch_bounds__(256, 1)
void segsum_kernel(const float* __restrict__ x, const int* __restrict__ idx,
                   float* __restrict__ out_ea_raw, float* __restrict__ cnt) {
  const int t = threadIdx.x;
  const int k = blockIdx.x * 64 + (t & 63);
  const int c = t >> 6;
  float acc[64];
#pragma unroll
  for (int j = 0; j < 64; ++j) acc[j] = 0.f;
  float cn = 0.f;
  for (int n = 0; n < N_TOK; ++n) {        // fixed n order -> deterministic fp sum
    if (idx[n] == k) {
      const float* xp = x + (size_t)n * D_DIM + c * 64;
#pragma unroll
      for (int j = 0; j < 64; ++j) acc[j] += xp[j];
      cn += 1.f;
    }
  }
#pragma unroll
  for (int j = 0; j < 64; ++j)
    out_ea_raw[(size_t)(c * 64 + j) * K_TOK + k] = acc[j];
  if (c == 0) cnt[k] = cn;
}

__global__ void ncs_kernel(const float* __restrict__ cs, const float* __restrict__ cnt,
                           float* __restrict__ out_ncs) {
  int k = blockIdx.x * 256 + threadIdx.x;
  out_ncs[k] = DECAY * cs[k] + ONEMD * cnt[k];
}

__global__ void nsum_kernel(const float* __restrict__ ncs, float* __restrict__ nsum) {
  __shared__ float sm[256];
  const int t = threadIdx.x;
  float s = 0.f;
  for (int i = t; i < K_TOK; i += 256) s += ncs[i];
  sm[t] = s;
  __syncthreads();
  for (int k = 128; k > 0; k >>= 1) {
    if (t < k) sm[t] += sm[t + k];
    __syncthreads();
  }
  if (t == 0) nsum[0] = sm[0];
}

__global__ void finalize_kernel(const float* __restrict__ ea_in, const float* __restrict__ ncs,
                                const float* __restrict__ nsum, float* __restrict__ out_embed,
                                float* __restrict__ out_ea) {
  const size_t o = (size_t)blockIdx.x * 256 + threadIdx.x;
  const int k = (int)(o & (K_TOK - 1));
  const float raw = out_ea[o];                       // raw embed_sum written by segsum
  const float ea  = DECAY * ea_in[o] + ONEMD * raw;  // new_embed_avg
  out_ea[o] = ea;
  const float nn  = nsum[0];
  const float csk = (ncs[k] + EPSV) / (nn + (float)K_TOK * EPSV) * nn;
  out_embed[o] = ea / csk;                           // new_embed
}

extern "C" void kernel_launch(void* const* d_in, const int* in_sizes, int n_in,
                              void* d_out, int out_size, void* d_ws, size_t ws_size,
                              hipStream_t stream) {
  (void)in_sizes; (void)n_in; (void)out_size; (void)ws_size;
  const float* in_x     = (const float*)d_in[0];   // [16,32,32,256]
  const float* in_embed = (const float*)d_in[1];   // [256,8192]
  const float* in_cs    = (const float*)d_in[2];   // [8192]
  const float* in_ea    = (const float*)d_in[3];   // [256,8192]

  float* out      = (float*)d_out;
  float* out_main = out;                       // 4194304  [16,256,32,32]
  float* out_diff = out + OUT_MAIN_SZ;         // 1
  float* out_ind  = out + OUT_MAIN_SZ + 1;     // 16384
  float* out_emb  = out + 4210689;             // 2097152  new_embed
  float* out_ncs  = out + 6307841;             // 8192     new_cluster_size
  float* out_ea   = out + 6316033;             // 2097152  new_embed_avg

  float* ws    = (float*)d_ws;
  float* ws_e2 = ws;                            // 8192
  float* ws_pv = ws + 8192;                     // KCHUNKS*16384 = 65536
  int*   ws_pi = (int*)(ws + 73728);            // 65536
  int*   ws_ix = (int*)(ws + 139264);           // 16384
  float* ws_ct = ws + 155648;                   // 8192
  float* ws_dp = ws + 163840;                   // 16384
  float* ws_n  = ws + 180224;                   // 1

  e2_kernel       <<<K_TOK / 256, 256, 0, stream>>>(in_embed, ws_e2);
  vq_argmin_kernel<<<dim3(128, KCHUNKS), 256, 0, stream>>>(in_x, in_embed, ws_e2, ws_pv, ws_pi);
  combine_kernel  <<<N_TOK / 256, 256, 0, stream>>>(ws_pv, ws_pi, ws_ix);
  ind_kernel      <<<N_TOK / 256, 256, 0, stream>>>(ws_ix, out_ind);
  quant_out_kernel<<<OUT_MAIN_SZ / 256, 256, 0, stream>>>(in_x, in_embed, ws_ix, out_main, ws_dp);
  diff_reduce_kernel<<<1, 256, 0, stream>>>(ws_dp, out_diff);
  segsum_kernel   <<<128,         256, 0, stream>>>(in_x, ws_ix, out_ea, ws_ct);
  ncs_kernel      <<<K_TOK / 256, 256, 0, stream>>>(in_cs, ws_ct, out_ncs);
  nsum_kernel     <<<1, 256, 0, stream>>>(out_ncs, ws_n);
  finalize_kernel <<<(D_DIM * K_TOK) / 256, 256, 0, stream>>>(in_ea, out_ncs, ws_n, out_emb, out_ea);
}